// RoIHeads_74947179315737
// MI455X (gfx1250) — compile-verified
//
#include <hip/hip_runtime.h>
#include <hip/hip_bf16.h>
#include <math.h>

typedef __attribute__((ext_vector_type(2))) float v2f;
typedef __attribute__((ext_vector_type(8))) float v8f;

#define B_IMGS    8
#define P_PROP    2000
#define C_CLS     81
#define M_CAND    160000      // P_PROP * (C_CLS-1)
#define K_PRE     2048
#define DET_N     100
#define MASK_W    64          // K_PRE / 32
#define W_IMGF    1333.0f
#define H_IMGF    800.0f
#define SCORE_T   0.05f
#define NMS_T     0.5f
#define BBOX_CLIP 4.135166556742356f   // log(1000/16)
#define OFF_SCALE 1334.0f              // max(W,H)+1

// ---------------------------------------------------------------- decode ----
__device__ __forceinline__ bool decode_clip(const float* r4,
                                            float px1, float py1, float px2, float py2,
                                            float& x1, float& y1, float& x2, float& y2) {
  float w  = px2 - px1, h = py2 - py1;
  float cx = px1 + 0.5f * w, cy = py1 + 0.5f * h;
  float dx = r4[0] / 10.0f;
  float dy = r4[1] / 10.0f;
  float dw = fminf(r4[2] / 5.0f, BBOX_CLIP);
  float dh = fminf(r4[3] / 5.0f, BBOX_CLIP);
  float pcx = dx * w + cx, pcy = dy * h + cy;
  float pw  = expf(dw) * w, ph = expf(dh) * h;
  x1 = fminf(fmaxf(pcx - 0.5f * pw, 0.0f), W_IMGF);
  y1 = fminf(fmaxf(pcy - 0.5f * ph, 0.0f), H_IMGF);
  x2 = fminf(fmaxf(pcx + 0.5f * pw, 0.0f), W_IMGF);
  y2 = fminf(fmaxf(pcy + 0.5f * ph, 0.0f), H_IMGF);
  return ((x2 - x1) >= 0.01f) && ((y2 - y1) >= 0.01f);
}

// ------------------------------------------------------- bitonic (desc) -----
template <int N, int TPB>
__device__ void bitonic_desc(unsigned long long* keys, int tid) {
  for (int k = 2; k <= N; k <<= 1) {
    for (int j = k >> 1; j > 0; j >>= 1) {
      __syncthreads();
      for (int i = tid; i < N; i += TPB) {
        int l = i ^ j;
        if (l > i) {
          unsigned long long a = keys[i], c = keys[l];
          bool up = ((i & k) == 0);
          bool sw = up ? (a < c) : (a > c);   // descending overall
          if (sw) { keys[i] = c; keys[l] = a; }
        }
      }
    }
  }
  __syncthreads();
}

// ---------------------------------------------- K1: softmax + masked score --
__global__ __launch_bounds__(256)
void k_scores(const float* __restrict__ logits, const float* __restrict__ boxreg,
              const float* __restrict__ props, float* __restrict__ scores) {
  int t = blockIdx.x * blockDim.x + threadIdx.x;
  if (t >= B_IMGS * P_PROP) return;
  int b = t / P_PROP, p = t % P_PROP;
  const float* lg = logits + (size_t)t * C_CLS;
  const float* rr = boxreg + (size_t)t * (C_CLS * 4);
  __builtin_prefetch(rr, 0, 0);          // global_prefetch_b8
  float mx = -3.4e38f;
  for (int c = 0; c < C_CLS; ++c) mx = fmaxf(mx, lg[c]);
  float denom = 0.0f;
  for (int c = 0; c < C_CLS; ++c) denom += expf(lg[c] - mx);
  float inv = 1.0f / denom;
  float px1 = props[t * 4 + 0], py1 = props[t * 4 + 1];
  float px2 = props[t * 4 + 2], py2 = props[t * 4 + 3];
  float* out = scores + (size_t)b * M_CAND + (size_t)p * (C_CLS - 1);
  for (int c = 1; c < C_CLS; ++c) {
    float s = expf(lg[c] - mx) * inv;
    float m = -1.0f;
    if (s > SCORE_T) {
      float x1, y1, x2, y2;
      if (decode_clip(rr + c * 4, px1, py1, px2, py2, x1, y1, x2, y2)) m = s;
    }
    out[c - 1] = m;
  }
}

// --------------------------------- K2: per-image top-2048 (sorted, stable) --
__global__ __launch_bounds__(1024)
void k_topk(const float* __restrict__ scores, int* __restrict__ sel_idx,
            float* __restrict__ sel_score) {
  __shared__ unsigned int hist[4096];
  __shared__ unsigned long long keys[4096];
  __shared__ unsigned int cnt, binCut;
  int b = blockIdx.x, tid = threadIdx.x;
  const float* sc = scores + (size_t)b * M_CAND;
  for (int i = tid; i < 4096; i += 1024) hist[i] = 0;
  __syncthreads();
  for (int m = tid; m < M_CAND; m += 1024) {
    float s = sc[m];
    if (s > 0.0f) {
      int bin = (int)(s * 4096.0f); if (bin > 4095) bin = 4095;
      atomicAdd(&hist[bin], 1u);
    }
  }
  __syncthreads();
  for (int i = tid; i < 4096; i += 1024) keys[i] = 0ull;
  if (tid == 0) {
    unsigned total = 0; int bc = 0;
    for (int bb = 4095; bb >= 0; --bb) {
      total += hist[bb];
      if (total >= K_PRE) { bc = bb; break; }
    }
    binCut = (unsigned)bc; cnt = 0;
  }
  __syncthreads();
  unsigned bcut = binCut;
  for (int m = tid; m < M_CAND; m += 1024) {
    float s = sc[m];
    if (s > 0.0f) {
      int bin = (int)(s * 4096.0f); if (bin > 4095) bin = 4095;
      if ((unsigned)bin >= bcut) {
        unsigned pos = atomicAdd(&cnt, 1u);
        if (pos < 4096u)
          keys[pos] = ((unsigned long long)__float_as_uint(s) << 32) |
                      (unsigned long long)(0xFFFFFFFFu - (unsigned)m);
      }
    }
  }
  bitonic_desc<4096, 1024>(keys, tid);
  for (int k = tid; k < K_PRE; k += 1024) {
    unsigned long long key = keys[k];
    if (key == 0ull) { sel_idx[b * K_PRE + k] = -1; sel_score[b * K_PRE + k] = -1.0f; }
    else {
      sel_idx[b * K_PRE + k]   = (int)(0xFFFFFFFFu - (unsigned)key);
      sel_score[b * K_PRE + k] = __uint_as_float((unsigned)(key >> 32));
    }
  }
}

// --------------------------------------------- K3: decode selected boxes ----
__global__ __launch_bounds__(256)
void k_decode_sel(const float* __restrict__ boxreg, const float* __restrict__ props,
                  const int* __restrict__ sel_idx,
                  float4* __restrict__ top_b, int* __restrict__ top_l,
                  float4* __restrict__ boxoff, float* __restrict__ area) {
  int t = blockIdx.x * blockDim.x + threadIdx.x;
  if (t >= B_IMGS * K_PRE) return;
  int b = t >> 11;
  int idx = sel_idx[t];
  float4 tb, bo; int lb; float ar;
  if (idx >= 0) {
    int p = idx / (C_CLS - 1);
    int c = idx % (C_CLS - 1) + 1;
    int row = b * P_PROP + p;
    const float* pr = props + (size_t)row * 4;
    const float* r4 = boxreg + (size_t)row * (C_CLS * 4) + c * 4;
    float x1, y1, x2, y2;
    decode_clip(r4, pr[0], pr[1], pr[2], pr[3], x1, y1, x2, y2);
    tb = make_float4(x1, y1, x2, y2);
    lb = c;
    float off = (float)c * OFF_SCALE;
    bo = make_float4(x1 + off, y1 + off, x2 + off, y2 + off);
    ar = (x2 - x1) * (y2 - y1);
  } else {
    tb = make_float4(0.f, 0.f, 0.f, 0.f);
    lb = 0;
    bo = make_float4(-1e8f, -1e8f, -1e8f, -1e8f);
    ar = 0.0f;
  }
  top_b[t] = tb; top_l[t] = lb; boxoff[t] = bo; area[t] = ar;
}

// ------------- K4: suppression bitmask; WMMA computes area_i+area_j tile ----
__global__ __launch_bounds__(128)
void k_mask(const float4* __restrict__ boxoff, const float* __restrict__ area,
            unsigned int* __restrict__ mask) {
  __shared__ float4 sbox[K_PRE];
  __shared__ float  sarea[K_PRE];
  int blk = blockIdx.x;
  int b  = blk >> 7;          // 128 i-tiles per image
  int it = blk & 127;
  for (int k = threadIdx.x; k < K_PRE; k += 128) {
    sbox[k]  = boxoff[b * K_PRE + k];
    sarea[k] = area[b * K_PRE + k];
  }
  __syncthreads();
  int wave = threadIdx.x >> 5;
  int lane = threadIdx.x & 31;
  int i0 = it * 16;
  int iGlob = i0 + (lane & 15);
  float4 bi = sbox[iGlob];
  // B operand (N = i rows of this tile): B[0][n]=1, B[1][n]=area_i(n); K=2,3 zero.
  float bm = (lane < 16) ? 1.0f : 0.0f;
  v2f Bop; Bop[0] = bm; Bop[1] = sarea[iGlob] * bm;
  for (int jw = wave; jw < MASK_W; jw += 4) {
    unsigned int word = 0;
#pragma unroll
    for (int half = 0; half < 2; ++half) {
      int j0 = jw * 32 + half * 16;
      // A operand (M = j cols): A[m][0]=area_j(m), A[m][1]=1; K=2,3 zero.
      float am = (lane < 16) ? 1.0f : 0.0f;
      v2f Aop; Aop[0] = sarea[j0 + (lane & 15)] * am; Aop[1] = am;
      v8f cc = {};
      // D[m][n] = area_j(j0+m) + area_i(i0+n)  (rank-2 matmul)
      v8f d = __builtin_amdgcn_wmma_f32_16x16x4_f32(
          false, Aop, false, Bop, (short)0, cc, false, false);
      unsigned int bits = 0;
      int jbase = j0 + ((lane < 16) ? 0 : 8);
#pragma unroll
      for (int r = 0; r < 8; ++r) {
        int j = jbase + r;
        float4 bj = sbox[j];
        float iw = fminf(bi.z, bj.z) - fmaxf(bi.x, bj.x);
        float ih = fminf(bi.w, bj.w) - fmaxf(bi.y, bj.y);
        float inter = fmaxf(iw, 0.0f) * fmaxf(ih, 0.0f);
        float uni = fmaxf(d[r] - inter, 1e-9f);
        bool sup = (inter > NMS_T * uni) && (j > iGlob);
        bits |= (sup ? (1u << r) : 0u);
      }
      unsigned int other = __shfl_xor(bits, 16);       // wave32 half-pair merge
      unsigned int w16 = (lane < 16) ? (bits | (other << 8)) : 0u;
      word |= w16 << (16 * half);
    }
    if (lane < 16)
      mask[((size_t)b * K_PRE + (size_t)iGlob) * MASK_W + jw] = word;
  }
}

// ----------------------------- K5: exact sequential greedy scan (1 wave) ----
__global__ __launch_bounds__(32)
void k_scan(const unsigned int* __restrict__ mask, const float* __restrict__ sel_score,
            float* __restrict__ finals) {
  __shared__ unsigned int remv[64];
  int b = blockIdx.x, lane = threadIdx.x;
  remv[lane] = 0; remv[lane + 32] = 0;
  __syncthreads();
  const unsigned int* mrow = mask + (size_t)b * K_PRE * MASK_W;
  for (int i = 0; i < K_PRE; ++i) {
    float s = sel_score[b * K_PRE + i];
    unsigned int rv = remv[i >> 5];
    bool kept = (s > 0.0f) && !((rv >> (i & 31)) & 1u);
    if (kept) {
      remv[lane]      |= mrow[(size_t)i * MASK_W + lane];
      remv[lane + 32] |= mrow[(size_t)i * MASK_W + 32 + lane];
    }
    if (lane == 0) finals[b * K_PRE + i] = kept ? s : -1.0f;
  }
}

// ----------------------------------- K6: top-100 + gather + write outputs ---
__global__ __launch_bounds__(256)
void k_final(const float* __restrict__ finals, const float4* __restrict__ top_b,
             const int* __restrict__ top_l, float* __restrict__ out) {
  __shared__ unsigned long long keys[K_PRE];
  int b = blockIdx.x, tid = threadIdx.x;
  for (int k = tid; k < K_PRE; k += 256) {
    float s = finals[b * K_PRE + k];
    unsigned int hi = (s > 0.0f) ? __float_as_uint(s) : 0u;
    keys[k] = ((unsigned long long)hi << 32) |
              (unsigned long long)(0xFFFFFFFFu - (unsigned)k);
  }
  bitonic_desc<K_PRE, 256>(keys, tid);
  for (int d = tid; d < DET_N; d += 256) {
    unsigned long long key = keys[d];
    int k = (int)(0xFFFFFFFFu - (unsigned)key);
    unsigned int hi = (unsigned)(key >> 32);
    float s = hi ? __uint_as_float(hi) : -1.0f;
    float4 bx = top_b[b * K_PRE + k];
    int lb = top_l[b * K_PRE + k];
    float* ob = out + (size_t)b * DET_N * 4 + (size_t)d * 4;       // det_b
    ob[0] = bx.x; ob[1] = bx.y; ob[2] = bx.z; ob[3] = bx.w;
    out[B_IMGS * DET_N * 4 + b * DET_N + d]     = s;               // det_s
    out[B_IMGS * DET_N * 5 + b * DET_N + d]     = (float)lb;       // det_l
    out[B_IMGS * DET_N * 6 + b * DET_N + d]     = (s > 0.0f) ? 1.0f : 0.0f; // det_valid
  }
}

// ---------------------------------------------------------------- launch ----
extern "C" void kernel_launch(void* const* d_in, const int* in_sizes, int n_in,
                              void* d_out, int out_size, void* d_ws, size_t ws_size,
                              hipStream_t stream) {
  (void)in_sizes; (void)n_in; (void)out_size; (void)ws_size;
  const float* logits = (const float*)d_in[0];   // [16000, 81]
  const float* boxreg = (const float*)d_in[1];   // [16000, 324]
  const float* props  = (const float*)d_in[2];   // [8, 2000, 4]
  float* out = (float*)d_out;

  size_t off = 0;
  auto sub = [&](size_t bytes) {
    void* p = (char*)d_ws + off;
    off += (bytes + 255) & ~(size_t)255;
    return p;
  };
  float*        scores    = (float*)sub(sizeof(float) * B_IMGS * M_CAND);
  int*          sel_idx   = (int*)sub(sizeof(int) * B_IMGS * K_PRE);
  float*        sel_score = (float*)sub(sizeof(float) * B_IMGS * K_PRE);
  float4*       top_b     = (float4*)sub(sizeof(float4) * B_IMGS * K_PRE);
  int*          top_l     = (int*)sub(sizeof(int) * B_IMGS * K_PRE);
  float4*       boxoff    = (float4*)sub(sizeof(float4) * B_IMGS * K_PRE);
  float*        area      = (float*)sub(sizeof(float) * B_IMGS * K_PRE);
  unsigned int* maskbuf   = (unsigned int*)sub(sizeof(unsigned int) * B_IMGS * K_PRE * MASK_W);
  float*        finals    = (float*)sub(sizeof(float) * B_IMGS * K_PRE);

  k_scores<<<(B_IMGS * P_PROP + 255) / 256, 256, 0, stream>>>(logits, boxreg, props, scores);
  k_topk<<<B_IMGS, 1024, 0, stream>>>(scores, sel_idx, sel_score);
  k_decode_sel<<<(B_IMGS * K_PRE + 255) / 256, 256, 0, stream>>>(
      boxreg, props, sel_idx, top_b, top_l, boxoff, area);
  k_mask<<<B_IMGS * 128, 128, 0, stream>>>(boxoff, area, maskbuf);
  k_scan<<<B_IMGS, 32, 0, stream>>>(maskbuf, sel_score, finals);
  k_final<<<B_IMGS, 256, 0, stream>>>(finals, top_b, top_l, out);
}